// forwadWarp_87505663689322
// MI455X (gfx1250) — compile-verified
//
#include <hip/hip_runtime.h>

// Problem shape from the reference: N=8, C=16, H=256, W=448
#define N_ 8
#define C_ 16
#define H_ 256
#define W_ 448
#define HW_ (H_ * W_)

// Block tile: 64 wide (w) x 4 tall (h) = 256 threads = 8 waves (wave32).
#define TW 64
#define TH 4

static_assert(W_ % TW == 0, "W must tile by TW");
static_assert(H_ % TH == 0, "H must tile by TH");

// ---------------------------------------------------------------------------
// Stage 1: scatter (forward warp). One thread per (n,h,w); flow/corner math is
// computed once and amortized over the 16 channels (flow broadcasts over C).
// acc lives in d_out, norm lives in d_ws; both pre-zeroed by memset.
// Atomics are no-return f32 adds -> global_atomic_add_f32 (STOREcnt), which
// resolve in the 192MB L2 since acc+norm (117MB) fit entirely.
// Grid is 3D (W/TW, H/TH, N) so no integer div/mod is needed for indexing.
// ---------------------------------------------------------------------------
__global__ __launch_bounds__(TW * TH) void splat_kernel(
    const float* __restrict__ src,
    const float* __restrict__ flow,
    const float* __restrict__ weight,
    float* __restrict__ acc,
    float* __restrict__ nrm)
{
    const int w  = blockIdx.x * TW + threadIdx.x;
    const int h  = blockIdx.y * TH + threadIdx.y;
    const int n  = blockIdx.z;
    const int hw = h * W_ + w;

    // flow: (N, 2, H, W); broadcast over channels
    const float fx = flow[((size_t)n * 2 + 0) * HW_ + hw];
    const float fy = flow[((size_t)n * 2 + 1) * HW_ + hw];

    const float xx = (float)w + fx;
    const float yy = (float)h + fy;
    const float xf = floorf(xx);
    const float yf = floorf(yy);
    const float a  = xx - xf;   // alpha
    const float b  = yy - yf;   // beta

    // 4 bilinear corners: LT, RT, LB, RB (matches reference order)
    const float xc[4] = { xf, xf + 1.0f, xf,        xf + 1.0f };
    const float yc[4] = { yf, yf,        yf + 1.0f, yf + 1.0f };
    float bw[4];
    bw[0] = (1.0f - a) * (1.0f - b);
    bw[1] = a * (1.0f - b);
    bw[2] = (1.0f - a) * b;
    bw[3] = a * b;

    bool ok[4];
    int  offs[4];
#pragma unroll
    for (int k = 0; k < 4; ++k) {
        // float-domain bounds test, exactly as the reference does
        ok[k] = (xc[k] >= 0.0f) && (xc[k] < (float)W_) &&
                (yc[k] >= 0.0f) && (yc[k] < (float)H_);
        offs[k] = ok[k] ? ((int)yc[k] * W_ + (int)xc[k]) : 0;
    }

    const size_t nbase = (size_t)n * C_ * HW_ + (size_t)hw;
    const float* sp = src    + nbase;
    const float* wp = weight + nbase;

#pragma unroll 4
    for (int c = 0; c < C_; ++c) {
        // prefetch two channel planes ahead (gfx1250: global_prefetch_b8)
        if (c + 2 < C_) {
            __builtin_prefetch(sp + (size_t)(c + 2) * HW_, 0, 0);
            __builtin_prefetch(wp + (size_t)(c + 2) * HW_, 0, 0);
        }
        const float s  = sp[(size_t)c * HW_];
        const float wt = wp[(size_t)c * HW_];
        const float sw = s * wt;   // src*weight, like reference's `src`
        const size_t plane = ((size_t)n * C_ + c) * HW_;
#pragma unroll
        for (int k = 0; k < 4; ++k) {
            if (ok[k]) {
                // fire-and-forget hardware f32 atomic adds (no return value)
                unsafeAtomicAdd(acc + plane + offs[k], sw * bw[k]);
                unsafeAtomicAdd(nrm + plane + offs[k], wt * bw[k]);
            }
        }
    }
}

// ---------------------------------------------------------------------------
// Stage 2: normalize in place. out = (norm != 0) ? acc/norm : 0
// float4-vectorized; total element count (14,680,064) is divisible by 4.
// ---------------------------------------------------------------------------
__global__ __launch_bounds__(256) void normalize_kernel(
    float4* __restrict__ acc4,
    const float4* __restrict__ nrm4,
    int nvec)
{
    const int i = blockIdx.x * blockDim.x + threadIdx.x;
    if (i >= nvec) return;
    float4 av = acc4[i];
    float4 sv = nrm4[i];
    float4 ov;
    ov.x = (sv.x != 0.0f) ? av.x / sv.x : 0.0f;
    ov.y = (sv.y != 0.0f) ? av.y / sv.y : 0.0f;
    ov.z = (sv.z != 0.0f) ? av.z / sv.z : 0.0f;
    ov.w = (sv.w != 0.0f) ? av.w / sv.w : 0.0f;
    acc4[i] = ov;
}

extern "C" void kernel_launch(void* const* d_in, const int* in_sizes, int n_in,
                              void* d_out, int out_size, void* d_ws, size_t ws_size,
                              hipStream_t stream) {
    const float* src    = (const float*)d_in[0];
    const float* flow   = (const float*)d_in[1];
    const float* weight = (const float*)d_in[2];

    float* acc = (float*)d_out;  // accumulator lives in the output buffer
    float* nrm = (float*)d_ws;   // normalizer lives in workspace

    const size_t total = (size_t)N_ * C_ * H_ * W_;

    // Zero accumulators every call (graph-capture-legal memset nodes).
    hipMemsetAsync(acc, 0, total * sizeof(float), stream);
    hipMemsetAsync(nrm, 0, total * sizeof(float), stream);

    dim3 block(TW, TH, 1);
    dim3 grid(W_ / TW, H_ / TH, N_);
    splat_kernel<<<grid, block, 0, stream>>>(src, flow, weight, acc, nrm);

    const int nvec = (int)(total / 4);
    normalize_kernel<<<(nvec + 255) / 256, 256, 0, stream>>>(
        (float4*)acc, (const float4*)nrm, nvec);
}